// WordGnn_54056458387640
// MI455X (gfx1250) — compile-verified
//
#include <hip/hip_runtime.h>

// -------- types for WMMA fragments (wave32, gfx1250) --------
typedef __attribute__((ext_vector_type(2))) float v2f;
typedef __attribute__((ext_vector_type(8))) float v8f;

#define FEAT 128  // EMBED == HIDDEN == 128

// ---------------------------------------------------------------------------
// 1) zero a float buffer (degree accumulators must be re-zeroed every call)
// ---------------------------------------------------------------------------
__global__ void k_zero_f32(float* __restrict__ p, int n) {
    int i = blockIdx.x * blockDim.x + threadIdx.x;
    if (i < n) p[i] = 0.0f;
}

// ---------------------------------------------------------------------------
// 2) weighted out/in degree via global f32 atomics
// ---------------------------------------------------------------------------
__global__ void k_degrees(const int* __restrict__ src, const int* __restrict__ dst,
                          const float* __restrict__ w,
                          float* __restrict__ outDeg, float* __restrict__ inDeg,
                          int nEdges) {
    int e = blockIdx.x * blockDim.x + threadIdx.x;
    if (e >= nEdges) return;
    float we = w[e];
    atomicAdd(outDeg + src[e], we);
    atomicAdd(inDeg + dst[e], we);
}

// ---------------------------------------------------------------------------
// 3) nw[e] = w[e] * rsqrt(outDeg[src]) * rsqrt(inDeg[dst])
// ---------------------------------------------------------------------------
__global__ void k_edge_norm(const int* __restrict__ src, const int* __restrict__ dst,
                            const float* __restrict__ w,
                            const float* __restrict__ outDeg, const float* __restrict__ inDeg,
                            float* __restrict__ nw, int nEdges) {
    int e = blockIdx.x * blockDim.x + threadIdx.x;
    if (e >= nEdges) return;
    nw[e] = w[e] * rsqrtf(outDeg[src[e]]) * rsqrtf(inDeg[dst[e]]);
}

// ---------------------------------------------------------------------------
// 4) Y = X @ W  via V_WMMA_F32_16X16X4_F32 (exact fp32, matches reference).
//    Block = 256 threads = 8 waves; W (64KB) staged once per block into LDS
//    in a K-pair-interleaved, bank-swizzled layout:
//       sW[k2*256 + ((2*col + (k&1)) ^ ((k2&1)<<5))] = W[k][col],  k2 = k>>1
//    so each B fragment {W[krow][c], W[krow+1][c]} is ONE aligned ds_load_b64
//    (krow = k + 2*hi is even), and lo/hi half-waves hit complementary
//    32-bank halves -> conflict-free.
//    Each wave owns a 16-row x 128-col strip: 8 v8f accumulators, 32 k-steps.
// ---------------------------------------------------------------------------
__global__ void k_gemm_wmma(const float* __restrict__ X, const float* __restrict__ W,
                            float* __restrict__ Y, int nRows) {
    __shared__ float sW[FEAT * FEAT];  // 64 KB of the 320 KB WGP LDS

    const int tid = threadIdx.x;
    // cooperative staging with interleave + swizzle
    {
        const float4* Wv = (const float4*)W;
        for (int i = tid; i < FEAT * FEAT / 4; i += 256) {
            float4 v  = Wv[i];
            int k     = i >> 5;          // row of W
            int col   = (i & 31) << 2;   // starting col of this float4
            int k2    = k >> 1;
            int p     = k & 1;
            int sw    = (k2 & 1) << 5;
            int base  = k2 * (2 * FEAT);
            sW[base + (((2 * (col + 0)) + p) ^ sw)] = v.x;
            sW[base + (((2 * (col + 1)) + p) ^ sw)] = v.y;
            sW[base + (((2 * (col + 2)) + p) ^ sw)] = v.z;
            sW[base + (((2 * (col + 3)) + p) ^ sw)] = v.w;
        }
    }
    __syncthreads();

    const int wave = tid >> 5;
    const int lane = tid & 31;
    const int lo   = lane & 15;
    const int hi   = lane >> 4;

    const int tile = blockIdx.x * 8 + wave;          // 16-row tile index
    if (tile * 16 >= nRows) return;                  // wave-uniform: EXEC stays full

    const float* Xrow = X + (size_t)(tile * 16 + lo) * FEAT;

    v8f acc[8];
    const v8f zero = {0.f, 0.f, 0.f, 0.f, 0.f, 0.f, 0.f, 0.f};
#pragma unroll
    for (int j = 0; j < 8; ++j) acc[j] = zero;

    for (int k = 0; k < FEAT; k += 4) {
        const int krow = k + 2 * hi;                 // even
        const int k2l  = krow >> 1;                  // = k/2 + hi
        const int sw   = (k2l & 1) << 5;             // == hi<<5 (k/2 is even)
        // A fragment: contiguous float2 -> coalesced global_load_b64
        const v2f a = *(const v2f*)(Xrow + krow);
        const float* sWrow = sW + (size_t)k2l * (2 * FEAT);
#pragma unroll
        for (int j = 0; j < 8; ++j) {
            // one aligned ds_load_b64 per B fragment, bank-conflict-free
            const v2f b = *(const v2f*)(sWrow + ((2 * (j * 16 + lo)) ^ sw));
            acc[j] = __builtin_amdgcn_wmma_f32_16x16x4_f32(
                false, a, false, b, (short)0, acc[j], false, false);
        }
    }

    // C/D layout: VGPR r -> row (r + 8*hi), col = j*16 + lo
    float* Yt = Y + (size_t)tile * 16 * FEAT;
#pragma unroll
    for (int j = 0; j < 8; ++j)
#pragma unroll
        for (int r = 0; r < 8; ++r)
            Yt[(size_t)(r + 8 * hi) * FEAT + j * 16 + lo] = acc[j][r];
}

// ---------------------------------------------------------------------------
// 5) agg[v][:] = b[:]  (bias broadcast init, float4 stores)
// ---------------------------------------------------------------------------
__global__ void k_bias_init(float* __restrict__ out, const float* __restrict__ b, int nRows) {
    int i = blockIdx.x * blockDim.x + threadIdx.x;   // over nRows * 32 float4 slots
    if (i >= nRows * (FEAT / 4)) return;
    ((float4*)out)[i] = ((const float4*)b)[i & (FEAT / 4 - 1)];
}

// ---------------------------------------------------------------------------
// 6) scatter-add: one wave per edge, lane f handles features 4f..4f+3
//    gather h[src] (global_load_b128) * nw[e], atomic add into out[dst]
// ---------------------------------------------------------------------------
__global__ void k_edge_agg(const int* __restrict__ src, const int* __restrict__ dst,
                           const float* __restrict__ nw, const float* __restrict__ h,
                           float* __restrict__ out, int nEdges) {
    const int e    = (int)((blockIdx.x * (size_t)blockDim.x + threadIdx.x) >> 5);
    const int lane = threadIdx.x & 31;
    if (e >= nEdges) return;                         // wave-uniform exit
    const int   s   = src[e];
    const int   d   = dst[e];
    const float wgt = nw[e];
    const float4 hv = *(const float4*)(h + (size_t)s * FEAT + lane * 4);
    float* o = out + (size_t)d * FEAT + lane * 4;
    atomicAdd(o + 0, hv.x * wgt);
    atomicAdd(o + 1, hv.y * wgt);
    atomicAdd(o + 2, hv.z * wgt);
    atomicAdd(o + 3, hv.w * wgt);
}

// ---------------------------------------------------------------------------
// host-side launch
// inputs: 0=src[E] i32, 1=dst[E] i32, 2=w[E] f32, 3=emb[N*128] f32,
//         4=W1[128*128] f32, 5=b1[128] f32, 6=W2[128*128] f32, 7=b2[128] f32
// output: [N*128] f32
// ---------------------------------------------------------------------------
extern "C" void kernel_launch(void* const* d_in, const int* in_sizes, int n_in,
                              void* d_out, int out_size, void* d_ws, size_t ws_size,
                              hipStream_t stream) {
    const int*   src = (const int*)d_in[0];
    const int*   dst = (const int*)d_in[1];
    const float* w   = (const float*)d_in[2];
    const float* emb = (const float*)d_in[3];
    const float* W1  = (const float*)d_in[4];
    const float* b1  = (const float*)d_in[5];
    const float* W2  = (const float*)d_in[6];
    const float* b2  = (const float*)d_in[7];
    float*       out = (float*)d_out;

    const int E = in_sizes[0];
    const int N = in_sizes[3] / FEAT;

    // workspace layout (floats)
    float* ws     = (float*)d_ws;
    float* outDeg = ws;                                   // N
    float* inDeg  = ws + (size_t)N;                       // N
    float* nw     = ws + (size_t)2 * N;                   // E
    float* hbuf   = ws + (size_t)2 * N + E;               // N*128 (GEMM out, both layers)
    float* h1     = hbuf + (size_t)N * FEAT;              // N*128 (layer-1 aggregation)

    const int T = 256;

    // degrees + normalization
    k_zero_f32<<<(2 * N + T - 1) / T, T, 0, stream>>>(outDeg, 2 * N);
    k_degrees<<<(E + T - 1) / T, T, 0, stream>>>(src, dst, w, outDeg, inDeg, E);
    k_edge_norm<<<(E + T - 1) / T, T, 0, stream>>>(src, dst, w, outDeg, inDeg, nw, E);

    const int rowTiles   = (N + 15) / 16;                 // 6250
    const int gemmBlocks = (rowTiles + 7) / 8;            // 8 waves / block

    // ---- layer 1 ----
    k_gemm_wmma<<<gemmBlocks, 256, 0, stream>>>(emb, W1, hbuf, N);
    k_bias_init<<<((size_t)N * (FEAT / 4) + T - 1) / T, T, 0, stream>>>(h1, b1, N);
    k_edge_agg<<<(E + 7) / 8, T, 0, stream>>>(src, dst, nw, hbuf, h1, E);

    // ---- layer 2 ----
    k_gemm_wmma<<<gemmBlocks, 256, 0, stream>>>(h1, W2, hbuf, N);
    k_bias_init<<<((size_t)N * (FEAT / 4) + T - 1) / T, T, 0, stream>>>(out, b2, N);
    k_edge_agg<<<(E + 7) / 8, T, 0, stream>>>(src, dst, nw, hbuf, out, E);
}